// SignedDualGNN_42623255446220
// MI455X (gfx1250) — compile-verified
//
#include <hip/hip_runtime.h>
#include <cstdint>

#define NUM_USERS 100000
#define NUM_ITEMS 50000
#define N_NODES   150000
#define EMB_DIM   64
#define TILE_EDGES 256
#define TILES_PER_BLOCK 2

// ---------- d_out row mapping (outputs concatenated: i_users, d_users, i_items, d_items)
__device__ __forceinline__ long long out_row_i(int r) {
  return (r < NUM_USERS) ? (long long)r : (long long)(2 * NUM_USERS + (r - NUM_USERS));
}
__device__ __forceinline__ long long out_row_d(int r) {
  return (r < NUM_USERS) ? (long long)(NUM_USERS + r)
                         : (long long)(2 * NUM_USERS + NUM_ITEMS + (r - NUM_USERS));
}

__device__ __forceinline__ float4 f4_scale(float4 a, float s) {
  return make_float4(a.x * s, a.y * s, a.z * s, a.w * s);
}
__device__ __forceinline__ float4 f4_add(float4 a, float4 b) {
  return make_float4(a.x + b.x, a.y + b.y, a.z + b.z, a.w + b.w);
}

// ---------- CDNA5 async global->LDS copy (ASYNCcnt path), 16B per lane
__device__ __forceinline__ void async_copy_b128(unsigned lds_byte_off, const void* gaddr) {
  asm volatile("global_load_async_to_lds_b128 %0, %1, off"
               :: "v"(lds_byte_off), "v"(gaddr)
               : "memory");
}
__device__ __forceinline__ void wait_asynccnt0() {
  asm volatile("s_wait_asynccnt 0" ::: "memory");
}
__device__ __forceinline__ unsigned lds_off_of(const void* p) {
  return (unsigned)(unsigned long long)(uintptr_t)p;  // low 32 bits of LDS flat addr = LDS offset
}

// ---------- guaranteed-native no-return fp32 atomic add (STOREcnt-tracked;
// S_ENDPGM's implicit wait-idle covers completion at kernel end)
__device__ __forceinline__ void global_atomic_fadd(float* p, float v) {
  asm volatile("global_atomic_add_f32 %0, %1, off"
               :: "v"(p), "v"(v)
               : "memory");
}

// Stage one 256-edge tile (src/dst/val) into LDS. Full tiles: waves 0..2 each
// issue 2 async B128 copies per lane (32 lanes x 32B = 1KB per array).
__device__ __forceinline__ void stage_tile(
    int* ts, int* td, float* tv,
    const int* __restrict__ src, const int* __restrict__ dst,
    const float* __restrict__ val, int base, int n_edges, int tid) {
  const int count = n_edges - base;
  if (count >= TILE_EDGES) {
    const int wave = tid >> 5;
    const int lane = tid & 31;
    if (wave == 0) {
      unsigned l = lds_off_of(ts) + (unsigned)lane * 32u;
      const char* g = (const char*)(src + base) + lane * 32;
      async_copy_b128(l, g);
      async_copy_b128(l + 16u, g + 16);
    } else if (wave == 1) {
      unsigned l = lds_off_of(td) + (unsigned)lane * 32u;
      const char* g = (const char*)(dst + base) + lane * 32;
      async_copy_b128(l, g);
      async_copy_b128(l + 16u, g + 16);
    } else if (wave == 2) {
      unsigned l = lds_off_of(tv) + (unsigned)lane * 32u;
      const char* g = (const char*)(val + base) + lane * 32;
      async_copy_b128(l, g);
      async_copy_b128(l + 16u, g + 16);
    }
  } else {
    if (tid < count) {   // partial last tile: plain loads (DScnt-ordered by barrier)
      ts[tid] = src[base + tid];
      td[tid] = dst[base + tid];
      tv[tid] = val[base + tid];
    }
  }
}

// ---------- init: cur = concat(user, item); seed the layer-0 sum term into d_out
__global__ __launch_bounds__(256) void init_kernel(
    const float4* __restrict__ ui, const float4* __restrict__ ii,
    const float4* __restrict__ ud, const float4* __restrict__ id,
    float4* __restrict__ cur_i, float4* __restrict__ cur_d, float4* __restrict__ out) {
  long long idx = (long long)blockIdx.x * blockDim.x + threadIdx.x;
  const long long total = (long long)N_NODES * (EMB_DIM / 4);
  if (idx >= total) return;
  int node = (int)(idx >> 4);
  int sub  = (int)(idx & 15);
  float4 vi, vd;
  if (node < NUM_USERS) {
    vi = ui[idx]; vd = ud[idx];
  } else {
    long long j = idx - (long long)NUM_USERS * 16;
    vi = ii[j];  vd = id[j];
  }
  cur_i[idx] = vi;
  cur_d[idx] = vd;
  out[out_row_i(node) * 16 + sub] = vi;
  out[out_row_d(node) * 16 + sub] = vd;
}

// ---------- layer-0 residual: nxt = (1+eps)*deg_inv[node]*cur
__global__ __launch_bounds__(256) void residual_kernel(
    const float4* __restrict__ ci, const float4* __restrict__ cd,
    float4* __restrict__ ni, float4* __restrict__ nd,
    const float* __restrict__ deg_pos, const float* __restrict__ deg_neg,
    const float* __restrict__ eps_pos, const float* __restrict__ eps_neg, int layer) {
  long long idx = (long long)blockIdx.x * blockDim.x + threadIdx.x;
  const long long total = (long long)N_NODES * (EMB_DIM / 4);
  if (idx >= total) return;
  int node = (int)(idx >> 4);
  float sp = (1.0f + eps_pos[layer]) * deg_pos[node];
  float sn = (1.0f + eps_neg[layer]) * deg_neg[node];
  ni[idx] = f4_scale(ci[idx], sp);
  nd[idx] = f4_scale(cd[idx], sn);
}

// ---------- fused: out += layer result; nxt = (1+eps)*deg_inv*layer result
__global__ __launch_bounds__(256) void accum_residual_kernel(
    const float4* __restrict__ ci, const float4* __restrict__ cd,
    float4* __restrict__ ni, float4* __restrict__ nd, float4* __restrict__ out,
    const float* __restrict__ deg_pos, const float* __restrict__ deg_neg,
    const float* __restrict__ eps_pos, const float* __restrict__ eps_neg, int layer) {
  long long idx = (long long)blockIdx.x * blockDim.x + threadIdx.x;
  const long long total = (long long)N_NODES * (EMB_DIM / 4);
  if (idx >= total) return;
  int node = (int)(idx >> 4);
  int sub  = (int)(idx & 15);
  float4 a = ci[idx];
  float4 b = cd[idx];
  long long oi = out_row_i(node) * 16 + sub;
  long long od = out_row_d(node) * 16 + sub;
  out[oi] = f4_add(out[oi], a);
  out[od] = f4_add(out[od], b);
  float sp = (1.0f + eps_pos[layer]) * deg_pos[node];
  float sn = (1.0f + eps_neg[layer]) * deg_neg[node];
  ni[idx] = f4_scale(a, sp);
  nd[idx] = f4_scale(b, sn);
}

// ---------- SpMM scatter with double-buffered async edge staging.
// 16 lanes per edge; grid-stride over 256-edge tiles; next tile's async
// global->LDS copies overlap with current tile's gather+atomic work.
__global__ __launch_bounds__(256) void spmm_scatter_kernel(
    const float* __restrict__ x, float* __restrict__ out,
    const int* __restrict__ src, const int* __restrict__ dst,
    const float* __restrict__ val, int n_edges) {
  __shared__ int   s_src[2][TILE_EDGES];
  __shared__ int   s_dst[2][TILE_EDGES];
  __shared__ float s_val[2][TILE_EDGES];

  const int tid    = threadIdx.x;
  const int nTiles = (n_edges + TILE_EDGES - 1) / TILE_EDGES;
  int tile = blockIdx.x;
  int buf  = 0;

  if (tile < nTiles)
    stage_tile(s_src[0], s_dst[0], s_val[0], src, dst, val,
               tile * TILE_EDGES, n_edges, tid);

  const int lane16 = tid & 15;
  const float4* __restrict__ x4 = (const float4*)x;

  for (; tile < nTiles; tile += gridDim.x) {
    if (tid < 96) wait_asynccnt0();   // waves 0..2 drain their async copies
    __syncthreads();                  // LDS tile visible to all 8 waves

    const int nextTile = tile + gridDim.x;
    if (nextTile < nTiles)            // overlap next tile's copy with compute
      stage_tile(s_src[buf ^ 1], s_dst[buf ^ 1], s_val[buf ^ 1],
                 src, dst, val, nextTile * TILE_EDGES, n_edges, tid);

    const int base = tile * TILE_EDGES;
    int count = n_edges - base;
    if (count > TILE_EDGES) count = TILE_EDGES;
    const int*   bs = s_src[buf];
    const int*   bd = s_dst[buf];
    const float* bv = s_val[buf];

    for (int e = tid >> 4; e < count; e += 16) {
      const int en = e + 16;
      if (en < count)                 // L2 prefetch of next gather row
        __builtin_prefetch(&x4[(size_t)bd[en] * 16 + lane16], 0, 0);
      const int   s = bs[e];
      const int   d = bd[e];
      const float v = bv[e];
      float4 xv = x4[(size_t)d * 16 + lane16];        // coalesced 256B row gather
      float* o = out + (size_t)s * EMB_DIM + lane16 * 4;
      global_atomic_fadd(o + 0, v * xv.x);            // native global_atomic_add_f32
      global_atomic_fadd(o + 1, v * xv.y);
      global_atomic_fadd(o + 2, v * xv.z);
      global_atomic_fadd(o + 3, v * xv.w);
    }
    buf ^= 1;
    // next iteration's wait+barrier provides the buffer-reuse sync
  }
}

// ---------- final: out = (out + last layer result) / NUM_LAYERS
__global__ __launch_bounds__(256) void final_kernel(
    const float4* __restrict__ ni, const float4* __restrict__ nd, float4* __restrict__ out) {
  long long idx = (long long)blockIdx.x * blockDim.x + threadIdx.x;
  const long long total = (long long)N_NODES * (EMB_DIM / 4);
  if (idx >= total) return;
  int node = (int)(idx >> 4);
  int sub  = (int)(idx & 15);
  long long oi = out_row_i(node) * 16 + sub;
  long long od = out_row_d(node) * 16 + sub;
  out[oi] = f4_scale(f4_add(out[oi], ni[idx]), 1.0f / 3.0f);
  out[od] = f4_scale(f4_add(out[od], nd[idx]), 1.0f / 3.0f);
}

extern "C" void kernel_launch(void* const* d_in, const int* in_sizes, int n_in,
                              void* d_out, int out_size, void* d_ws, size_t ws_size,
                              hipStream_t stream) {
  const float* user_i  = (const float*)d_in[0];
  const float* item_i  = (const float*)d_in[1];
  const float* user_d  = (const float*)d_in[2];
  const float* item_d  = (const float*)d_in[3];
  const float* eps_pos = (const float*)d_in[4];
  const float* eps_neg = (const float*)d_in[5];
  const int*   pos_src = (const int*)d_in[6];
  const int*   pos_dst = (const int*)d_in[7];
  const float* pos_val = (const float*)d_in[8];
  const float* pos_deg = (const float*)d_in[9];
  const int*   neg_src = (const int*)d_in[10];
  const int*   neg_dst = (const int*)d_in[11];
  const float* neg_val = (const float*)d_in[12];
  const float* neg_deg = (const float*)d_in[13];
  const int    n_edges = in_sizes[6];

  float* out = (float*)d_out;
  const size_t nodeElems = (size_t)N_NODES * EMB_DIM;
  float* A_i = (float*)d_ws;
  float* B_i = A_i + nodeElems;
  float* A_d = B_i + nodeElems;
  float* B_d = A_d + nodeElems;

  const int threads = 256;
  const long long nodeVec = (long long)N_NODES * (EMB_DIM / 4);
  const int nodeBlocks = (int)((nodeVec + threads - 1) / threads);
  const int nTiles     = (n_edges + TILE_EDGES - 1) / TILE_EDGES;
  const int edgeBlocks = (nTiles + TILES_PER_BLOCK - 1) / TILES_PER_BLOCK;

  // x0 -> A; out = x0
  init_kernel<<<nodeBlocks, threads, 0, stream>>>(
      (const float4*)user_i, (const float4*)item_i,
      (const float4*)user_d, (const float4*)item_d,
      (float4*)A_i, (float4*)A_d, (float4*)out);

  // layer 0: B = (1+eps0)*deg*A + SpMM(A)   (= x1)
  residual_kernel<<<nodeBlocks, threads, 0, stream>>>(
      (const float4*)A_i, (const float4*)A_d, (float4*)B_i, (float4*)B_d,
      pos_deg, neg_deg, eps_pos, eps_neg, 0);
  spmm_scatter_kernel<<<edgeBlocks, threads, 0, stream>>>(A_i, B_i, pos_src, pos_dst, pos_val, n_edges);
  spmm_scatter_kernel<<<edgeBlocks, threads, 0, stream>>>(A_d, B_d, neg_src, neg_dst, neg_val, n_edges);

  // out += x1; A = (1+eps1)*deg*B  (residual part of layer 1)
  accum_residual_kernel<<<nodeBlocks, threads, 0, stream>>>(
      (const float4*)B_i, (const float4*)B_d, (float4*)A_i, (float4*)A_d, (float4*)out,
      pos_deg, neg_deg, eps_pos, eps_neg, 1);

  // layer 1: A += SpMM(B)   (= x2)
  spmm_scatter_kernel<<<edgeBlocks, threads, 0, stream>>>(B_i, A_i, pos_src, pos_dst, pos_val, n_edges);
  spmm_scatter_kernel<<<edgeBlocks, threads, 0, stream>>>(B_d, A_d, neg_src, neg_dst, neg_val, n_edges);

  // out = (out + x2) / 3
  final_kernel<<<nodeBlocks, threads, 0, stream>>>(
      (const float4*)A_i, (const float4*)A_d, (float4*)out);
}